// LTCCell_87660282511867
// MI455X (gfx1250) — compile-verified
//
#include <hip/hip_runtime.h>
#include <stdint.h>

// LTC cell, B=512, I=128, N=256, 6 ODE unfolds.
// Compute-bound (fp32 VALU + v_tanh_f32 transcendentals); params (~1.5MB) are
// L2-resident (192MB L2, 23.3TB/s HBM is irrelevant here).
// WMMA is NOT applicable: the j-contraction's summand sigmoid(sigma*(v-mu)) is
// non-separable in (b,j,n) -> no bilinear AxB form. CDNA5-specific paths used
// instead: wave32 codegen, LDS state tile, async global->LDS double-buffered
// parameter prefetch (global_load_async_to_lds_b32 / s_wait_asynccnt).

#define B_TOT    512
#define I_DIM    128
#define N_DIM    256
#define STEPS    6
#define BT       4          // batch rows per workgroup -> 128 workgroups
#define NTHREADS 256        // 8 wave32s; thread tid owns output column n=tid

#define AS1 __attribute__((address_space(1)))
#define AS3 __attribute__((address_space(3)))

#if __has_builtin(__builtin_amdgcn_global_load_async_to_lds_b32) && \
    __has_builtin(__builtin_amdgcn_s_wait_asynccnt)
#define HAVE_ASYNC 1
#endif

#if __has_builtin(__builtin_amdgcn_tanhf)
#define HAVE_TANH 1
#endif

#ifdef HAVE_ASYNC
__device__ __forceinline__ void async_ld_f32(const float* g, float* l) {
  // Builtin is declared with typed int* params (b32 payload).
  // Flat->AS cast via integer: low 32 bits of a flat LDS address are the LDS
  // byte address (ISA 10.2 aperture rules); flat global == global address.
  __builtin_amdgcn_global_load_async_to_lds_b32(
      (AS1 int*)(uintptr_t)g, (AS3 int*)(uintptr_t)l, 0, 0);
}
#endif

// One parameter row (fixed i-or-j, this thread's column n) accumulated into
// num/den for BT batch rows. Math: w = W*sigmoid(sigma*(x-mu)); den+=w; num+=w*erev.
__device__ __forceinline__ void gate_accum(float muv, float sgv, float wv, float ev,
                                           const float* x, int xstride,
                                           float* num, float* den) {
#ifdef HAVE_TANH
  // sigmoid(t) = 0.5 + 0.5*tanh(t/2);  w = hW*(1+tanh(a*x+c)), a=sigma/2
  const float a  = 0.5f * sgv;
  const float c  = -a * muv;
  const float hw = 0.5f * wv;
  #pragma unroll
  for (int b = 0; b < BT; ++b) {
    float th = __builtin_amdgcn_tanhf(fmaf(a, x[b * xstride], c));  // v_tanh_f32
    float w  = fmaf(hw, th, hw);
    den[b] += w;
    num[b]  = fmaf(w, ev, num[b]);   // erev is +-1.0f
  }
#else
  // sigmoid(t) = rcp(1 + exp2(-log2e * t))
  const float a = -1.4426950408889634f * sgv;
  const float c = -a * muv;
  #pragma unroll
  for (int b = 0; b < BT; ++b) {
    float e = __builtin_amdgcn_exp2f(fmaf(a, x[b * xstride], c));   // v_exp_f32
    float s = __builtin_amdgcn_rcpf(1.0f + e);                      // v_rcp_f32
    float w = wv * s;
    den[b] += w;
    num[b]  = fmaf(w, ev, num[b]);
  }
#endif
}

// Sum of gated rows 0..ROWS-1 (ROWS even). Async path: software-pipelined,
// double-buffered via LDS, unrolled x2 so buffer indices are compile-time
// constants (immediate-offset ds loads, no per-row branch).
// Each thread async-copies exactly the 4 LDS words it will read
// (pbuf[buf][a][tid]), so per-wave ASYNCcnt ordering suffices: no barriers.
template <int ROWS>
__device__ __forceinline__ void gated_row_sum(
    const float* __restrict__ p_mu, const float* __restrict__ p_sg,
    const float* __restrict__ p_w,  const float* __restrict__ p_ev,
    const float* x_lds, int xstride, int tid,
#ifdef HAVE_ASYNC
    float (*pbuf)[N_DIM],            // [8][N_DIM] == [2 bufs][4 arrays][N]
#endif
    float* num, float* den)
{
#ifdef HAVE_ASYNC
  auto issue = [&](int r, int buf) {
    async_ld_f32(&p_mu[r * N_DIM + tid], &pbuf[buf * 4 + 0][tid]);
    async_ld_f32(&p_sg[r * N_DIM + tid], &pbuf[buf * 4 + 1][tid]);
    async_ld_f32(&p_w [r * N_DIM + tid], &pbuf[buf * 4 + 2][tid]);
    async_ld_f32(&p_ev[r * N_DIM + tid], &pbuf[buf * 4 + 3][tid]);
  };
  issue(0, 0);
  for (int j = 0; j < ROWS; j += 2) {
    issue(j + 1, 1);                       // j+1 <= ROWS-1 always (ROWS even)
    __builtin_amdgcn_s_wait_asynccnt(4);   // <=8 in flight; row j complete
    gate_accum(pbuf[0][tid], pbuf[1][tid], pbuf[2][tid], pbuf[3][tid],
               &x_lds[j], xstride, num, den);
    if (j + 2 < ROWS) {
      issue(j + 2, 0);                     // buf0 already consumed above
      __builtin_amdgcn_s_wait_asynccnt(4); // row j+1 complete
    } else {
      __builtin_amdgcn_s_wait_asynccnt(0); // drain tail
    }
    gate_accum(pbuf[4][tid], pbuf[5][tid], pbuf[6][tid], pbuf[7][tid],
               &x_lds[j + 1], xstride, num, den);
  }
#else
  for (int j = 0; j < ROWS; ++j) {
    gate_accum(p_mu[j * N_DIM + tid], p_sg[j * N_DIM + tid],
               p_w [j * N_DIM + tid], p_ev[j * N_DIM + tid],
               &x_lds[j], xstride, num, den);
  }
#endif
}

__global__ __launch_bounds__(NTHREADS) void ltc_cell_kernel(
    const float* __restrict__ inputs,  // (B,I)
    const float* __restrict__ state,   // (B,N)
    const float* __restrict__ smu,     // (I,N)
    const float* __restrict__ ssig,    // (I,N)
    const float* __restrict__ sW,      // (I,N)
    const float* __restrict__ serev,   // (I,N)
    const float* __restrict__ mu,      // (N,N)
    const float* __restrict__ sigma,   // (N,N)
    const float* __restrict__ W,       // (N,N)
    const float* __restrict__ erev,    // (N,N)
    const float* __restrict__ vleak,   // (N)
    const float* __restrict__ gleak,   // (N)
    const float* __restrict__ cm_t,    // (N)
    float* __restrict__ out)           // (2,B,N) : tuple (v, v)
{
  __shared__ float v_lds[BT * N_DIM];    // 4 KB : v_pre tile
  __shared__ float in_lds[BT * I_DIM];   // 2 KB : input tile
#ifdef HAVE_ASYNC
  __shared__ float pbuf[8][N_DIM];       // 8 KB : 2 bufs x 4 param rows
#endif

  const int tid = threadIdx.x;           // n column
  const int b0  = blockIdx.x * BT;

  for (int idx = tid; idx < BT * N_DIM; idx += NTHREADS)
    v_lds[idx] = state[b0 * N_DIM + idx];
  for (int idx = tid; idx < BT * I_DIM; idx += NTHREADS)
    in_lds[idx] = inputs[b0 * I_DIM + idx];

  const float cm      = cm_t[tid];
  const float gl      = gleak[tid];
  const float gl_vl   = gl * vleak[tid];
  const float denom_c = cm + gl;

  __syncthreads();

  // ---- sensory gating: computed once, reused by all 6 unfolds ----
  float sens_num[BT], sens_den[BT];
  #pragma unroll
  for (int b = 0; b < BT; ++b) { sens_num[b] = 0.0f; sens_den[b] = 0.0f; }

  gated_row_sum<I_DIM>(smu, ssig, sW, serev, in_lds, I_DIM, tid,
#ifdef HAVE_ASYNC
                       pbuf,
#endif
                       sens_num, sens_den);

  // ---- 6 ODE unfolds ----
  float vnew[BT];
  for (int s = 0; s < STEPS; ++s) {
    float num[BT], den[BT];
    #pragma unroll
    for (int b = 0; b < BT; ++b) { num[b] = sens_num[b]; den[b] = sens_den[b]; }

    gated_row_sum<N_DIM>(mu, sigma, W, erev, v_lds, N_DIM, tid,
#ifdef HAVE_ASYNC
                         pbuf,
#endif
                         num, den);

    #pragma unroll
    for (int b = 0; b < BT; ++b) {
      float vpre  = v_lds[b * N_DIM + tid];
      float numer = fmaf(cm, vpre, gl_vl + num[b]);
      float denom = denom_c + den[b];
      vnew[b] = numer / denom;             // outer loop only: exact division
    }
    __syncthreads();
    #pragma unroll
    for (int b = 0; b < BT; ++b) v_lds[b * N_DIM + tid] = vnew[b];
    __syncthreads();
  }

  // tuple (v, v): two concatenated copies
  #pragma unroll
  for (int b = 0; b < BT; ++b) {
    const int row = b0 + b;
    out[row * N_DIM + tid]                 = vnew[b];
    out[B_TOT * N_DIM + row * N_DIM + tid] = vnew[b];
  }
}

extern "C" void kernel_launch(void* const* d_in, const int* in_sizes, int n_in,
                              void* d_out, int out_size, void* d_ws, size_t ws_size,
                              hipStream_t stream) {
  (void)in_sizes; (void)n_in; (void)out_size; (void)d_ws; (void)ws_size;
  const float* inputs = (const float*)d_in[0];
  const float* state  = (const float*)d_in[1];
  const float* smu    = (const float*)d_in[2];
  const float* ssig   = (const float*)d_in[3];
  const float* sW     = (const float*)d_in[4];
  const float* serev  = (const float*)d_in[5];
  const float* mu     = (const float*)d_in[6];
  const float* sigma  = (const float*)d_in[7];
  const float* W      = (const float*)d_in[8];
  const float* erev   = (const float*)d_in[9];
  const float* vleak  = (const float*)d_in[10];
  const float* gleak  = (const float*)d_in[11];
  const float* cm     = (const float*)d_in[12];

  ltc_cell_kernel<<<B_TOT / BT, NTHREADS, 0, stream>>>(
      inputs, state, smu, ssig, sW, serev, mu, sigma, W, erev,
      vleak, gleak, cm, (float*)d_out);
}